// MoEGate_66391604461902
// MI455X (gfx1250) — compile-verified
//
#include <hip/hip_runtime.h>
#include <hip/hip_bf16.h>

typedef __attribute__((ext_vector_type(2))) float v2f;
typedef __attribute__((ext_vector_type(8))) float v8f;

#define B_   4
#define T_   4096
#define C_   2048
#define E_   64
#define TOPK 6
#define NTOK (B_ * T_)
#define ALPHA_ 0.001f
#define ROUTED_SCALING_ 1.0f

#define KC 64          // K chunk staged in LDS per iteration
#define XS 68          // padded LDS row stride (floats): 68 mod 64 = 4 -> conflict-free fragment reads
#define LG 68          // padded logits row stride

// ---------------------------------------------------------------------------
// Kernel 0: zero the workspace accumulators (ws is poisoned by the harness).
// ---------------------------------------------------------------------------
__global__ void moe_zero_ws_kernel(float* __restrict__ ws) {
    int i = blockIdx.x * blockDim.x + threadIdx.x;
    if (i < 2 * B_ * E_) ws[i] = 0.0f;
}

// ---------------------------------------------------------------------------
// Kernel 1: gating GEMM (fp32 WMMA) + softmax + greedy top-6 + partial sums
//   block = 128 threads = 4 wave32; handles 16 tokens x 64 experts
//   wave w computes the 16x16 logits tile for experts [16w, 16w+16)
// ---------------------------------------------------------------------------
__global__ __launch_bounds__(128)
void moe_gate_kernel(const float* __restrict__ x,      // (NTOK, C)
                     const float* __restrict__ w,      // (E, C)
                     int*   __restrict__ idx_out,      // (NTOK, TOPK)
                     float* __restrict__ w_out,        // (NTOK, TOPK)
                     float* __restrict__ g_ssum,       // (B, E) score sums
                     float* __restrict__ g_cnt)        // (B, E) topk counts
{
    __shared__ float xs [16 * XS];   // x tile   16 x KC (padded)
    __shared__ float wsm[E_ * XS];   // w tile   64 x KC (padded)
    __shared__ float lg [16 * LG];   // logits   16 x 64 (padded)
    __shared__ float ssum[E_];
    __shared__ float cnt [E_];

    const int tid  = threadIdx.x;
    const int wave = tid >> 5;
    const int lane = tid & 31;
    const int tok0 = blockIdx.x * 16;

    // WMMA fragment coordinates (ISA 16x4 f32 A-layout / 4x16 B-layout):
    //   lane 0-15 : M/N = lane,     K-pair {0,1}
    //   lane16-31 : M/N = lane-16,  K-pair {2,3}
    const int mrow = lane & 15;
    const int kb   = (lane >> 4) * 2;
    const int e0   = wave * 16;

    v8f acc = {};

    const float* xbase = x + (size_t)tok0 * C_;

    for (int k0 = 0; k0 < C_; k0 += KC) {
        __syncthreads();   // previous iteration's fragment reads done before overwrite
        // stage x tile: 16*64 = 1024 floats, coalesced
        for (int i = tid; i < 16 * KC; i += 128) {
            int r = i >> 6, c = i & (KC - 1);
            xs[r * XS + c] = xbase[(size_t)r * C_ + k0 + c];
        }
        // stage weight tile: 64*64 = 4096 floats, coalesced (L2-resident, 512 KiB total)
        for (int i = tid; i < E_ * KC; i += 128) {
            int r = i >> 6, c = i & (KC - 1);
            wsm[r * XS + c] = w[(size_t)r * C_ + k0 + c];
        }
        // prefetch next x chunk (global_prefetch_b8)
        if (k0 + KC < C_) {
            int r = tid >> 3, c = (tid & 7) * 8;
            __builtin_prefetch(xbase + (size_t)r * C_ + k0 + KC + c, 0, 0);
        }
        __syncthreads();

        const float* ap = xs  + mrow * XS + kb;
        const float* bp = wsm + (e0 + mrow) * XS + kb;
        #pragma unroll
        for (int kk = 0; kk < KC; kk += 4) {
            v2f a  = *(const v2f*)(ap + kk);   // A[m][k..k+1]
            v2f bb = *(const v2f*)(bp + kk);   // B[k..k+1][n] == weight[n][k..k+1]
            acc = __builtin_amdgcn_wmma_f32_16x16x4_f32(
                      false, a, false, bb, (short)0, acc, false, false);
        }
    }

    // scatter C/D fragment (VGPR v -> rows v and v+8) into the logits tile
    {
        const int nn    = e0 + (lane & 15);
        const int mbase = 8 * (lane >> 4);
        #pragma unroll
        for (int v = 0; v < 8; ++v)
            lg[(mbase + v) * LG + nn] = acc[v];
    }
    if (tid < E_) { ssum[tid] = 0.0f; cnt[tid] = 0.0f; }
    __syncthreads();

    // softmax + greedy top-6, one thread per token (tiny: 16x64)
    if (tid < 16) {
        float* row = lg + tid * LG;
        float mx = -3.4e38f;
        for (int e = 0; e < E_; ++e) mx = fmaxf(mx, row[e]);
        float sum = 0.0f;
        for (int e = 0; e < E_; ++e) { float s = __expf(row[e] - mx); row[e] = s; sum += s; }
        const float inv = 1.0f / sum;
        // normalize in place; rotate the accumulation index so no two of the
        // 16 active lanes hit the same LDS-atomic address in the same step
        for (int off = 0; off < E_; ++off) {
            int e = (off + tid) & (E_ - 1);
            float s = row[e] * inv;
            row[e] = s;
            atomicAdd(&ssum[e], s);
        }
        const int tok = tok0 + tid;
        for (int k = 0; k < TOPK; ++k) {
            float bv = -1.0f; int bi = 0;
            for (int e = 0; e < E_; ++e) {
                float s = row[e];
                if (s > bv) { bv = s; bi = e; }
            }
            idx_out[tok * TOPK + k] = bi;
            w_out  [tok * TOPK + k] = bv * ROUTED_SCALING_;
            atomicAdd(&cnt[bi], 1.0f);
            row[bi] = -1.0f;
        }
    }
    __syncthreads();

    // one global atomic per (batch, expert) per block (blocks never straddle a batch)
    if (tid < E_) {
        const int b = tok0 / T_;
        atomicAdd(&g_ssum[b * E_ + tid], ssum[tid]);
        atomicAdd(&g_cnt [b * E_ + tid], cnt [tid]);
    }
}

// ---------------------------------------------------------------------------
// Kernel 2: aux_loss = mean_b( sum_e( ce * mean_scores ) ) * alpha
//   ce = cnt * E/(T*K) ; mean_scores = ssum / T ; B*E = 256 values
// ---------------------------------------------------------------------------
__global__ __launch_bounds__(256)
void moe_aux_kernel(const float* __restrict__ g_ssum,
                    const float* __restrict__ g_cnt,
                    float* __restrict__ aux_out)
{
    __shared__ float red[256];
    const int tid = threadIdx.x;
    const float ce = g_cnt[tid] * ((float)E_ / (float)(T_ * TOPK));
    const float ms = g_ssum[tid] * (1.0f / (float)T_);
    red[tid] = ce * ms;
    __syncthreads();
    for (int s = 128; s > 0; s >>= 1) {
        if (tid < s) red[tid] += red[tid + s];
        __syncthreads();
    }
    if (tid == 0) aux_out[0] = red[0] * (ALPHA_ / (float)B_);
}

// ---------------------------------------------------------------------------
extern "C" void kernel_launch(void* const* d_in, const int* in_sizes, int n_in,
                              void* d_out, int out_size, void* d_ws, size_t ws_size,
                              hipStream_t stream) {
    const float* x = (const float*)d_in[0];   // (B, T, C) fp32
    const float* w = (const float*)d_in[1];   // (E, C)    fp32

    // d_out layout (return order): topk_idx (NTOK*TOPK), topk_weight (NTOK*TOPK), aux_loss (1)
    int*   idx_out = (int*)d_out;
    float* w_out   = (float*)d_out + (size_t)NTOK * TOPK;
    float* aux_out = w_out + (size_t)NTOK * TOPK;

    float* g_ssum = (float*)d_ws;             // (B, E)
    float* g_cnt  = g_ssum + B_ * E_;         // (B, E)

    moe_zero_ws_kernel<<<1, 512, 0, stream>>>(g_ssum);
    moe_gate_kernel<<<NTOK / 16, 128, 0, stream>>>(x, w, idx_out, w_out, g_ssum, g_cnt);
    moe_aux_kernel<<<1, 256, 0, stream>>>(g_ssum, g_cnt, aux_out);
}